// Render_47012712022039
// MI455X (gfx1250) — compile-verified
//
#include <hip/hip_runtime.h>

// ---------------------------------------------------------------------------
// Triangle rasterizer for MI455X (gfx1250, wave32).
// Edge functions are affine in (px,py)  =>  rank-2 matmul  =>  one
// V_WMMA_F32_16X16X4_F32 evaluates one edge over a full 16x16 pixel tile.
// ---------------------------------------------------------------------------

typedef __attribute__((ext_vector_type(2))) float v2f;
typedef __attribute__((ext_vector_type(8))) float v8f;

#define S 512
#define TILE 16
#define TILES_PER_DIM (S / TILE)                    // 32
#define NTILES (TILES_PER_DIM * TILES_PER_DIM)      // 1024
#define WAVES_PER_BLOCK 8
#define LIN_STEP (2.0f / 511.0f)                    // jnp.linspace(-1,1,512) step
#define EPS_W 1e-8f

struct TriSetup {
    float e[3][3];   // [edge AB,BC,CA][{B, C, A0}] -- each pre-multiplied by w
    float v[3][3];   // vertices [a,b,c][x,y,z]
    int   lo0, lo1, hi0, hi1;   // bbox in pixel indices; lo0==hi0 => culled
    int   pad0, pad1;           // -> 96 bytes / triangle
};

// --------------------------- kernel 1: zmin --------------------------------
__global__ __launch_bounds__(1024)
void zmin_kernel(const float* __restrict__ tris, int nverts,
                 float* __restrict__ ws_zmin, unsigned int* __restrict__ ws_zmax)
{
    __shared__ float red[1024];
    float m = 3.4e38f;
    for (int i = threadIdx.x; i < nverts; i += 1024)
        m = fminf(m, tris[i * 3 + 2]);
    red[threadIdx.x] = m;
    __syncthreads();
    for (int s = 512; s > 0; s >>= 1) {
        if (threadIdx.x < s)
            red[threadIdx.x] = fminf(red[threadIdx.x], red[threadIdx.x + s]);
        __syncthreads();
    }
    if (threadIdx.x == 0) { *ws_zmin = red[0]; *ws_zmax = 0u; }
}

// --------------------------- kernel 2: setup -------------------------------
__global__ __launch_bounds__(256)
void setup_kernel(const float* __restrict__ tris, int T, TriSetup* __restrict__ ts)
{
    int t = blockIdx.x * 256 + threadIdx.x;
    if (t >= T) return;
    const float* p = tris + t * 9;
    float ax = p[0], ay = p[1], az = p[2];
    float bx = p[3], by = p[4], bz = p[5];
    float cx = p[6], cy = p[7], cz = p[8];

    float w = (bx - ax) * (cy - ay) - (by - ay) * (cx - ax);
    bool wok = fabsf(w) > EPS_W;

    TriSetup s;
    // edge(p,q): area(px,py,p,q) = A0 + B*px + C*py with
    //   A0 = p0*q1 - p1*q0,  B = p1 - q1,  C = q0 - p0.   All scaled by w.
    s.e[0][0] = (ay - by) * w; s.e[0][1] = (bx - ax) * w; s.e[0][2] = (ax * by - ay * bx) * w; // AB
    s.e[1][0] = (by - cy) * w; s.e[1][1] = (cx - bx) * w; s.e[1][2] = (bx * cy - by * cx) * w; // BC
    s.e[2][0] = (cy - ay) * w; s.e[2][1] = (ax - cx) * w; s.e[2][2] = (cx * ay - cy * ax) * w; // CA
    s.v[0][0] = ax; s.v[0][1] = ay; s.v[0][2] = az;
    s.v[1][0] = bx; s.v[1][1] = by; s.v[1][2] = bz;
    s.v[2][0] = cx; s.v[2][1] = cy; s.v[2][2] = cz;

    float mnx = fminf(fminf(ax, bx), cx), mny = fminf(fminf(ay, by), cy);
    float mxx = fmaxf(fmaxf(ax, bx), cx), mxy = fmaxf(fmaxf(ay, by), cy);
    mnx = fminf(fmaxf(mnx, -1.0f), 1.0f); mny = fminf(fmaxf(mny, -1.0f), 1.0f);
    mxx = fminf(fmaxf(mxx, -1.0f), 1.0f); mxy = fminf(fmaxf(mxy, -1.0f), 1.0f);
    int lo0 = (int)floorf((mnx + 1.0f) * 0.5f * (float)S);
    int lo1 = (int)floorf((mny + 1.0f) * 0.5f * (float)S);
    int hi0 = (int)floorf((mxx + 1.0f) * 0.5f * (float)S);
    int hi1 = (int)floorf((mxy + 1.0f) * 0.5f * (float)S);
    if (!wok) { lo0 = 0; hi0 = 0; }   // degenerate: empty bbox -> always culled
    s.lo0 = lo0; s.lo1 = lo1; s.hi0 = hi0; s.hi1 = hi1;
    s.pad0 = 0; s.pad1 = 0;
    ts[t] = s;
}

// --------------------------- kernel 3: raster ------------------------------
// One wave32 per 16x16 tile; z-buffer and 4 output channels live in VGPRs.
__global__ __launch_bounds__(256)
void raster_kernel(const TriSetup* __restrict__ ts, int T,
                   const float* __restrict__ zmin_p,
                   unsigned int* __restrict__ zmax_bits,
                   float* __restrict__ out)
{
    const int lane = threadIdx.x & 31;
    // Force the tile id scalar so the bbox cull is an s_cbranch and EXEC is
    // all-ones at every WMMA (ISA requirement). It is uniform per wave anyway.
    const int tile = __builtin_amdgcn_readfirstlane(
        blockIdx.x * WAVES_PER_BLOCK + (threadIdx.x >> 5));
    const int tx0 = (tile >> 5) << 4;   // ix base of tile
    const int ty0 = (tile & 31) << 4;   // iy base of tile

    const float zmin = *zmin_p;
    const bool lo16 = lane < 16;
    const int  l15  = lane & 15;

    // A-matrix 16x4 (rank 2 used): row m = [px_m, 1, 0, 0]
    // lanes 0-15 hold K=0,1 for M=lane; lanes 16-31 hold K=2,3 (zero).
    v2f a_frag;
    a_frag.x = lo16 ? fmaf((float)(tx0 + lane), LIN_STEP, -1.0f) : 0.0f;
    a_frag.y = lo16 ? 1.0f : 0.0f;
    const float py_l = fmaf((float)(ty0 + lane), LIN_STEP, -1.0f); // N=lane (lanes 0-15)

    // D layout: lanes 0-15 -> (M=r, N=lane); lanes 16-31 -> (M=8+r, N=lane-16)
    const int ix_base = tx0 + (lo16 ? 0 : 8);
    const int iy      = ty0 + l15;

    float zb[8], rX[8], rY[8], rZ[8], rW[8];
#pragma unroll
    for (int r = 0; r < 8; ++r) { zb[r] = zmin; rX[r] = 0.0f; rY[r] = 0.0f; rZ[r] = 0.0f; rW[r] = 0.0f; }

    for (int t = 0; t < T; ++t) {
        const TriSetup* s = ts + t;           // uniform -> scalar loads
        const int lo0 = s->lo0, lo1 = s->lo1, hi0 = s->hi0, hi1 = s->hi1;
        if (hi0 <= tx0 || lo0 >= tx0 + TILE || hi1 <= ty0 || lo1 >= ty0 + TILE)
            continue;                          // scalar branch, EXEC untouched

        const v8f cz = {};
        v2f bf;
        // B-matrix 4x16: col n = [B*w, A0*w + C*w*py_n, 0, 0]
        bf.x = lo16 ? s->e[0][0] : 0.0f;
        bf.y = lo16 ? (s->e[0][2] + s->e[0][1] * py_l) : 0.0f;
        v8f pAB = __builtin_amdgcn_wmma_f32_16x16x4_f32(false, a_frag, false, bf,
                                                        (short)0, cz, false, false);
        bf.x = lo16 ? s->e[1][0] : 0.0f;
        bf.y = lo16 ? (s->e[1][2] + s->e[1][1] * py_l) : 0.0f;
        v8f pBC = __builtin_amdgcn_wmma_f32_16x16x4_f32(false, a_frag, false, bf,
                                                        (short)0, cz, false, false);
        bf.x = lo16 ? s->e[2][0] : 0.0f;
        bf.y = lo16 ? (s->e[2][2] + s->e[2][1] * py_l) : 0.0f;
        v8f pCA = __builtin_amdgcn_wmma_f32_16x16x4_f32(false, a_frag, false, bf,
                                                        (short)0, cz, false, false);

        const float v00 = s->v[0][0], v01 = s->v[0][1], v02 = s->v[0][2];
        const float v10 = s->v[1][0], v11 = s->v[1][1], v12 = s->v[1][2];
        const float v20 = s->v[2][0], v21 = s->v[2][1], v22 = s->v[2][2];

#pragma unroll
        for (int r = 0; r < 8; ++r) {
            float eAB = pAB[r], eBC = pBC[r], eCA = pCA[r];
            bool inside = (fmaxf(eAB, 0.0f) * fmaxf(eBC, 0.0f) * fmaxf(eCA, 0.0f)) > 0.0f;
            float safe = inside ? eAB : 1.0f;
            float inv  = 1.0f / safe;
            float bxv = eBC * inv;
            float byv = eCA * inv;
            float bzv = 1.0f - bxv - byv;
            float X = bxv * v00 + byv * v10 + bzv * v20;
            float Y = bxv * v01 + byv * v11 + bzv * v21;
            float Z = bxv * v02 + byv * v12 + bzv * v22;
            int ix = ix_base + r;
            bool bb = (ix >= lo0) && (ix < hi0) && (iy >= lo1) && (iy < hi1);
            bool m  = inside && bb && (Z >= zb[r]);
            zb[r] = m ? Z : zb[r];
            rX[r] = m ? X : rX[r];
            rY[r] = m ? Y : rY[r];
            rZ[r] = m ? Z : rZ[r];
            rW[r] = m ? 1.0f : rW[r];
        }
    }

    float tmax = 0.0f;
#pragma unroll
    for (int r = 0; r < 8; ++r) {
        float tz = zb[r] - zmin;              // >= 0 always
        tmax = fmaxf(tmax, tz);
        float* o = out + ((size_t)(ix_base + r) * S + iy) * 5;
        o[0] = rX[r]; o[1] = rY[r]; o[2] = rZ[r]; o[3] = rW[r]; o[4] = tz;
    }
    // non-negative floats: uint order == float order
    atomicMax(zmax_bits, __float_as_uint(tmax));
}

// --------------------------- kernel 4: normalize ---------------------------
__global__ __launch_bounds__(256)
void norm_kernel(float* __restrict__ out, const unsigned int* __restrict__ zmax_bits)
{
    int i = blockIdx.x * 256 + threadIdx.x;
    if (i >= S * S) return;
    float zmax = __uint_as_float(*zmax_bits);
    out[(size_t)i * 5 + 4] /= zmax;
}

// ---------------------------------------------------------------------------
extern "C" void kernel_launch(void* const* d_in, const int* in_sizes, int n_in,
                              void* d_out, int out_size, void* d_ws, size_t ws_size,
                              hipStream_t stream)
{
    (void)n_in; (void)out_size; (void)ws_size;
    const float* tris = (const float*)d_in[0];
    const int T = in_sizes[0] / 9;            // 2048 triangles (x,y,z)*3

    float*        ws_zmin = (float*)d_ws;
    unsigned int* ws_zmax = (unsigned int*)d_ws + 1;
    TriSetup*     ts      = (TriSetup*)((char*)d_ws + 256);
    float*        out     = (float*)d_out;

    zmin_kernel<<<1, 1024, 0, stream>>>(tris, T * 3, ws_zmin, ws_zmax);
    setup_kernel<<<(T + 255) / 256, 256, 0, stream>>>(tris, T, ts);
    raster_kernel<<<NTILES / WAVES_PER_BLOCK, 32 * WAVES_PER_BLOCK, 0, stream>>>(
        ts, T, ws_zmin, ws_zmax, out);
    norm_kernel<<<(S * S + 255) / 256, 256, 0, stream>>>(out, ws_zmax);
}